// GCNlearnable_72834055405955
// MI455X (gfx1250) — compile-verified
//
#include <hip/hip_runtime.h>

typedef __attribute__((ext_vector_type(2))) float v2f;
typedef __attribute__((ext_vector_type(4))) float v4f;
typedef __attribute__((ext_vector_type(8))) float v8f;

#define BM 256
#define BN 64
#define BK 32
#define LDA_S 265   // padded LDS stride (dwords) for A^T tile [BK][BM]; 265 coprime 64 banks
#define LDB_S 160   // padded LDS stride (dwords) for B pair tile [BK/2][BN][2]; 160%64==32

// C[b] = A[b] (MxK, row-major, lda) @ B[b] (KxN, row-major, ldb), fp32, WMMA f32 16x16x4.
// blockDim = 256 (8 waves). Block tile 256x64, wave tile 32x64 (2x4 subtiles of 16x16).
// B is staged K-pair-interleaved so each B fragment is one aligned ds_load_b64.
__launch_bounds__(256)
__global__ void gcn_gemm_f32_wmma(const float* __restrict__ A,
                                  const float* __restrict__ B,
                                  float* __restrict__ C,
                                  int K, int lda, int ldb, int ldc,
                                  long long sA, long long sB, long long sC)
{
    __shared__ float Asm[BK * LDA_S];          // A^T: Asm[k][m]
    __shared__ float Bsm[(BK / 2) * LDB_S];    // B pairs: Bsm[kp][2n+{0,1}] = B[2kp+{0,1}][n]

    const long long b = blockIdx.z;
    A += b * sA;
    B += b * sB;
    C += b * sC;

    const int m0 = blockIdx.x * BM;
    const int n0 = blockIdx.y * BN;

    const int t      = threadIdx.x;
    const int lane   = t & 31;
    const int wave   = t >> 5;        // 0..7, waves stacked along M
    const int lane15 = lane & 15;
    const int lhalf  = lane >> 4;     // 0 or 1 (K-half select per ISA layout)

    // A staging coordinates (float4 along K, transpose into LDS)
    const int a_c4 = (t & 7) * 4;     // k column (float4) within A tile
    const int a_r0 = t >> 3;          // 0..31, row base; 8 passes of 32 rows
    // B staging coordinates (float2 along N, pair rows 2kp/2kp+1)
    const int b_n2 = (t & 31) * 2;    // n column (float2) within B tile
    const int b_kp = t >> 5;          // 0..7 pair-row base; 2 passes of 8 pair-rows

    v8f acc[2][4] = {};

    const int kTiles = K / BK;
    for (int kt = 0; kt < kTiles; ++kt) {
        const int k0 = kt * BK;

        // ---- stage A tile (BM x BK), transposed into LDS ----
        #pragma unroll
        for (int r = 0; r < 8; ++r) {
            const int row = a_r0 + 32 * r;
            const v4f v = *(const v4f*)(A + (long long)(m0 + row) * lda + (k0 + a_c4));
            Asm[(a_c4 + 0) * LDA_S + row] = v.x;
            Asm[(a_c4 + 1) * LDA_S + row] = v.y;
            Asm[(a_c4 + 2) * LDA_S + row] = v.z;
            Asm[(a_c4 + 3) * LDA_S + row] = v.w;
        }
        // ---- stage B tile (BK x BN) as K-pairs: Bsm[kp][2n..2n+1] = {B[2kp][n], B[2kp+1][n]} ----
        #pragma unroll
        for (int r = 0; r < 2; ++r) {
            const int kp = b_kp + 8 * r;
            const v2f e = *(const v2f*)(B + (long long)(k0 + 2 * kp + 0) * ldb + (n0 + b_n2));
            const v2f o = *(const v2f*)(B + (long long)(k0 + 2 * kp + 1) * ldb + (n0 + b_n2));
            v4f pk;
            pk.x = e.x; pk.y = o.x;   // pair for column n2
            pk.z = e.y; pk.w = o.y;   // pair for column n2+1
            *(v4f*)&Bsm[kp * LDB_S + 2 * b_n2] = pk;   // 16B-aligned, full-width store
        }
        // prefetch next K tile into cache (global_prefetch_b8)
        if (kt + 1 < kTiles) {
            __builtin_prefetch(A + (long long)(m0 + a_r0) * lda + (k0 + BK + a_c4), 0, 0);
            __builtin_prefetch(B + (long long)(k0 + BK + 2 * b_kp) * ldb + (n0 + b_n2), 0, 0);
        }
        __syncthreads();

        // ---- 8 WMMA k-steps of K=4 ----
        #pragma unroll
        for (int ks = 0; ks < 8; ++ks) {
            const int kk = 4 * ks + 2 * lhalf;   // per-lane K pair select (A side)
            const int kp = 2 * ks + lhalf;       // pair-row select (B side)
            v2f af[2], bf[4];
            #pragma unroll
            for (int i = 0; i < 2; ++i) {
                const int m = wave * 32 + 16 * i + lane15;
                af[i].x = Asm[(kk + 0) * LDA_S + m];
                af[i].y = Asm[(kk + 1) * LDA_S + m];
            }
            #pragma unroll
            for (int j = 0; j < 4; ++j)
                bf[j] = *(const v2f*)&Bsm[kp * LDB_S + 2 * (16 * j + lane15)];
            #pragma unroll
            for (int i = 0; i < 2; ++i)
                #pragma unroll
                for (int j = 0; j < 4; ++j)
                    acc[i][j] = __builtin_amdgcn_wmma_f32_16x16x4_f32(
                        false, af[i], false, bf[j], (short)0, acc[i][j], false, false);
        }
        __syncthreads();
    }

    // ---- write out: C/D layout VGPR v -> rows v (lanes 0-15) and v+8 (lanes 16-31) ----
    #pragma unroll
    for (int i = 0; i < 2; ++i) {
        const int mrow = m0 + wave * 32 + 16 * i + 8 * lhalf;
        #pragma unroll
        for (int j = 0; j < 4; ++j) {
            float* dst = C + (long long)mrow * ldc + (n0 + 16 * j + lane15);
            #pragma unroll
            for (int v = 0; v < 8; ++v)
                dst[(long long)v * ldc] = acc[i][j][v];
        }
    }
}

extern "C" void kernel_launch(void* const* d_in, const int* in_sizes, int n_in,
                              void* d_out, int out_size, void* d_ws, size_t ws_size,
                              hipStream_t stream) {
    (void)in_sizes; (void)n_in; (void)out_size; (void)ws_size;
    const float* x    = (const float*)d_in[0];   // [32768, 256]
    // d_in[1] = batch ids (int64) — layout is contiguous blocks, not needed
    const float* W    = (const float*)d_in[2];   // [256, 256]
    const float* adj  = (const float*)d_in[3];   // [512, 512]
    float*       out  = (float*)d_out;           // [32768, 256]
    float*       h    = (float*)d_ws;            // intermediate x@W: 33.5 MB

    const int N_NODES = 512, N_BATCH = 64, D_IN = 256, D_OUT = 256;
    const int Mtot = N_BATCH * N_NODES;          // 32768

    // Pass 1: h = x @ W   (M=32768, N=256, K=256)
    dim3 g1(Mtot / BM, D_OUT / BN, 1);
    gcn_gemm_f32_wmma<<<g1, 256, 0, stream>>>(x, W, h,
                                              D_IN, D_IN, D_OUT, D_OUT,
                                              0LL, 0LL, 0LL);

    // Pass 2: out_b = adj @ h_b  (per batch: M=512, N=256, K=512; adj shared)
    dim3 g2(N_NODES / BM, D_OUT / BN, N_BATCH);
    gcn_gemm_f32_wmma<<<g2, 256, 0, stream>>>(adj, h, out,
                                              N_NODES, N_NODES, D_OUT, D_OUT,
                                              0LL,
                                              (long long)N_NODES * D_OUT,
                                              (long long)N_NODES * D_OUT);
}